// MultiHeadProbAttention_39144331936381
// MI455X (gfx1250) — compile-verified
//
#include <hip/hip_runtime.h>
#include <hip/hip_bf16.h>
#include <stdint.h>

// Problem constants (from reference)
#define B_SZ     8
#define L_SEQ    4096
#define DMODEL   1024
#define H_HEADS  16
#define D_HEAD   64
#define U_TOP    45
#define SKEYS    45

typedef __attribute__((ext_vector_type(2))) float v2f;
typedef __attribute__((ext_vector_type(8))) float v8f;

// ---------------- gfx1250 async global->LDS helpers ----------------
__device__ __forceinline__ void async_ld16(void* lds, const void* g) {
  unsigned l = (unsigned)(uintptr_t)lds;              // low 32 bits of shared aperture addr == LDS byte offset
  asm volatile("global_load_async_to_lds_b128 %0, %1, off"
               :: "v"(l), "v"(g) : "memory");
}
__device__ __forceinline__ void wait_async0() {
  asm volatile("s_wait_asynccnt 0" ::: "memory");
}

// ---------------- WMMA f32 GEMM: out = A(MxK) * W(NxK)^T + bias ----------------
// mode 0: out row-major [M,N]
// mode 1: out head-major [B,H,L,D_HEAD]  (n -> h,d ; m -> b,l)
#define MT  128
#define NT  64
#define KT  32
#define LDT 36   // padded LDS row stride in floats

__global__ __launch_bounds__(256) void k_gemm_wmma(
    const float* __restrict__ A, const float* __restrict__ W,
    const float* __restrict__ bias, float* __restrict__ out,
    int Kdim, int Ndim, int mode)
{
  __shared__ float sA[MT * LDT];
  __shared__ float sB[NT * LDT];
  const int t    = threadIdx.x;
  const int lane = t & 31;
  const int w    = t >> 5;        // wave id 0..7 -> 16 M-rows each
  const int half = lane >> 4;     // 0/1 (selects K pair / M+8)
  const int lidx = lane & 15;
  const int m0 = blockIdx.x * MT;
  const int n0 = blockIdx.y * NT;

  v8f acc[4];
  #pragma unroll
  for (int i = 0; i < 4; ++i)
    #pragma unroll
    for (int j = 0; j < 8; ++j) acc[i][j] = 0.0f;

  // staging assignments: A tile 128x32 (4 x b128/thread), B tile 64x32 (2 x b128/thread)
  const int ar = t >> 1;          // 0..127
  const int ac = (t & 1) * 16;    // 0/16
  const int br = t >> 2;          // 0..63
  const int bc = (t & 3) * 8;     // 0/8/16/24

  for (int k0 = 0; k0 < Kdim; k0 += KT) {
    const float* ga = A + (size_t)(m0 + ar) * Kdim + k0 + ac;
    #pragma unroll
    for (int i = 0; i < 4; ++i) async_ld16(&sA[ar * LDT + ac + i * 4], ga + i * 4);
    const float* gb = W + (size_t)(n0 + br) * Kdim + k0 + bc;
    #pragma unroll
    for (int i = 0; i < 2; ++i) async_ld16(&sB[br * LDT + bc + i * 4], gb + i * 4);
    wait_async0();
    __syncthreads();

    const int arow = (w * 16 + lidx) * LDT;
    #pragma unroll
    for (int kk = 0; kk < KT; kk += 4) {
      v2f a = *(const v2f*)&sA[arow + kk + 2 * half];
      #pragma unroll
      for (int nt = 0; nt < 4; ++nt) {
        v2f b = *(const v2f*)&sB[(nt * 16 + lidx) * LDT + kk + 2 * half];
        acc[nt] = __builtin_amdgcn_wmma_f32_16x16x4_f32(
            false, a, false, b, (short)0, acc[nt], false, false);
      }
    }
    __syncthreads();
  }

  #pragma unroll
  for (int nt = 0; nt < 4; ++nt) {
    int n = n0 + nt * 16 + lidx;
    float bv = bias[n];
    #pragma unroll
    for (int v = 0; v < 8; ++v) {
      int m = m0 + w * 16 + v + 8 * half;
      float val = acc[nt][v] + bv;
      if (mode == 0) {
        out[(size_t)m * Ndim + n] = val;
      } else {
        int b = m >> 12, l = m & (L_SEQ - 1);
        int h = n >> 6,  d = n & 63;
        out[(((size_t)b * H_HEADS + h) * L_SEQ + l) * D_HEAD + d] = val;
      }
    }
  }
}

// ---------------- sampled score stats: M = max_s(qk) - sum_s(qk)/L ----------------
__global__ __launch_bounds__(256) void k_sample(
    const float* __restrict__ Q, const float* __restrict__ K,
    const int* __restrict__ sidx, float* __restrict__ Mstat)
{
  int tg = blockIdx.x * 256 + threadIdx.x;
  if (tg >= B_SZ * H_HEADS * L_SEQ) return;
  int l  = tg & (L_SEQ - 1);
  int bh = tg >> 12;
  const float4* q = (const float4*)(Q + ((size_t)bh * L_SEQ + l) * D_HEAD);
  float4 qr[16];
  #pragma unroll
  for (int i = 0; i < 16; ++i) qr[i] = q[i];
  const float* Kb = K + (size_t)bh * L_SEQ * D_HEAD;
  float mx = -3.4e38f, sm = 0.0f;
  for (int s = 0; s < SKEYS; ++s) {
    int j = sidx[l * SKEYS + s];
    const float4* kr = (const float4*)(Kb + (size_t)j * D_HEAD);
    float acc = 0.0f;
    #pragma unroll
    for (int i = 0; i < 16; ++i) {
      float4 kv = kr[i];
      acc += qr[i].x * kv.x + qr[i].y * kv.y + qr[i].z * kv.z + qr[i].w * kv.w;
    }
    mx = fmaxf(mx, acc);
    sm += acc;
  }
  Mstat[tg] = mx - sm * (1.0f / (float)L_SEQ);
}

// ---------------- per-(b,h) iterative top-U (descending, lowest-index ties) ----------------
__global__ __launch_bounds__(256) void k_topk(
    const float* __restrict__ Mstat, int* __restrict__ Mtop)
{
  __shared__ float vals[L_SEQ];
  __shared__ float rv[256];
  __shared__ int   ri[256];
  int bh = blockIdx.x, t = threadIdx.x;
  for (int i = t; i < L_SEQ; i += 256) vals[i] = Mstat[(size_t)bh * L_SEQ + i];
  __syncthreads();
  for (int u = 0; u < U_TOP; ++u) {
    float bvv = -3.4e38f; int bii = 0;
    for (int i = t; i < L_SEQ; i += 256) {
      float v = vals[i];
      if (v > bvv) { bvv = v; bii = i; }
    }
    rv[t] = bvv; ri[t] = bii; __syncthreads();
    for (int s = 128; s > 0; s >>= 1) {
      if (t < s) {
        float ov = rv[t + s]; int oi = ri[t + s];
        if (ov > rv[t] || (ov == rv[t] && oi < ri[t])) { rv[t] = ov; ri[t] = oi; }
      }
      __syncthreads();
    }
    if (t == 0) { Mtop[bh * U_TOP + u] = ri[0]; vals[ri[0]] = -3.4e38f; }
    __syncthreads();
  }
}

// ---------------- dense attention for the U selected queries ----------------
__global__ __launch_bounds__(256) void k_attn(
    const float* __restrict__ Q, const float* __restrict__ K,
    const float* __restrict__ V, const int* __restrict__ Mtop,
    float* __restrict__ upd)
{
  __shared__ float sc[L_SEQ];
  __shared__ float qr[D_HEAD];
  __shared__ float red[256];
  int idx = blockIdx.x, bh = idx / U_TOP, u = idx % U_TOP;
  int t = threadIdx.x;
  int lq = Mtop[bh * U_TOP + u];
  if (t < D_HEAD) qr[t] = Q[((size_t)bh * L_SEQ + lq) * D_HEAD + t];
  __syncthreads();

  const float* Kb = K + (size_t)bh * L_SEQ * D_HEAD;
  float lmax = -3.4e38f;
  for (int k = t; k < L_SEQ; k += 256) {
    float s;
    if (k > lq) { s = -3.4e38f; }
    else {
      const float4* kr = (const float4*)(Kb + (size_t)k * D_HEAD);
      float acc = 0.0f;
      #pragma unroll
      for (int i = 0; i < 16; ++i) {
        float4 kv = kr[i];
        acc += qr[i*4] * kv.x + qr[i*4+1] * kv.y + qr[i*4+2] * kv.z + qr[i*4+3] * kv.w;
      }
      s = acc * 0.125f;   // 1/sqrt(64)
    }
    sc[k] = s;
    lmax = fmaxf(lmax, s);
  }
  red[t] = lmax; __syncthreads();
  for (int s = 128; s > 0; s >>= 1) { if (t < s) red[t] = fmaxf(red[t], red[t + s]); __syncthreads(); }
  float mx = red[0]; __syncthreads();

  float lsum = 0.0f;
  for (int k = t; k < L_SEQ; k += 256) { float e = __expf(sc[k] - mx); sc[k] = e; lsum += e; }
  red[t] = lsum; __syncthreads();
  for (int s = 128; s > 0; s >>= 1) { if (t < s) red[t] += red[t + s]; __syncthreads(); }
  float inv = 1.0f / red[0]; __syncthreads();

  int g = t >> 6, d = t & 63;
  const float* Vb = V + (size_t)bh * L_SEQ * D_HEAD;
  float accd = 0.0f;
  for (int k = g * 1024; k < (g + 1) * 1024; ++k) accd += sc[k] * Vb[(size_t)k * D_HEAD + d];
  red[t] = accd; __syncthreads();
  if (t < 64) {
    float r = (red[t] + red[t + 64] + red[t + 128] + red[t + 192]) * inv;
    upd[((size_t)bh * U_TOP + u) * D_HEAD + d] = r;
  }
}

// ---------------- cumsum(V) along L into flat [B, L, DMODEL] context ----------------
__global__ __launch_bounds__(256) void k_cumsum(
    const float* __restrict__ V, float* __restrict__ ctx)
{
  int t = blockIdx.x * 256 + threadIdx.x;   // B*DMODEL = 8192
  int b = t >> 10, c = t & 1023;
  int h = c >> 6,  d = c & 63;
  const float* vp = V + (((size_t)b * H_HEADS + h) * L_SEQ) * D_HEAD + d;
  float* cp = ctx + ((size_t)b * L_SEQ) * DMODEL + c;
  float acc = 0.0f;
  for (int l = 0; l < L_SEQ; ++l) {
    acc += vp[(size_t)l * D_HEAD];
    cp[(size_t)l * DMODEL] = acc;
  }
}

// ---------------- scatter updated rows into context ----------------
__global__ __launch_bounds__(256) void k_scatter(
    const int* __restrict__ Mtop, const float* __restrict__ upd,
    float* __restrict__ ctx)
{
  int t = blockIdx.x * 256 + threadIdx.x;
  if (t >= B_SZ * H_HEADS * U_TOP * D_HEAD) return;
  int d  = t & 63;
  int u  = (t >> 6) % U_TOP;
  int bh = t / (U_TOP * D_HEAD);
  int b = bh >> 4, h = bh & 15;
  int l = Mtop[bh * U_TOP + u];
  ctx[((size_t)b * L_SEQ + l) * DMODEL + h * D_HEAD + d] = upd[t];
}

// ---------------- host launcher ----------------
extern "C" void kernel_launch(void* const* d_in, const int* in_sizes, int n_in,
                              void* d_out, int out_size, void* d_ws, size_t ws_size,
                              hipStream_t stream)
{
  (void)in_sizes; (void)n_in; (void)out_size; (void)ws_size;
  const float* x    = (const float*)d_in[0];
  const int*   sidx = (const int*)  d_in[1];
  const float* Wq   = (const float*)d_in[2];
  const float* bq   = (const float*)d_in[3];
  const float* Wk   = (const float*)d_in[4];
  const float* bk   = (const float*)d_in[5];
  const float* Wv   = (const float*)d_in[6];
  const float* bv   = (const float*)d_in[7];
  const float* Wo   = (const float*)d_in[8];
  const float* bo   = (const float*)d_in[9];
  float* out = (float*)d_out;

  const size_t QKV = (size_t)B_SZ * H_HEADS * L_SEQ * D_HEAD; // 33,554,432 floats
  float* Qw    = (float*)d_ws;
  float* Kw    = Qw + QKV;
  float* Vw    = Kw + QKV;
  float* Mstat = Vw + QKV;                                    // B*H*L
  float* updw  = Mstat + (size_t)B_SZ * H_HEADS * L_SEQ;      // B*H*U*D_HEAD
  int*   Mtop  = (int*)(updw + (size_t)B_SZ * H_HEADS * U_TOP * D_HEAD);
  float* ctx   = Qw;  // Q is dead after k_attn; reuse its buffer for the context

  dim3 gridG((B_SZ * L_SEQ) / MT, DMODEL / NT);  // 256 x 16
  k_gemm_wmma<<<gridG, 256, 0, stream>>>(x, Wq, bq, Qw, DMODEL, DMODEL, 1);
  k_gemm_wmma<<<gridG, 256, 0, stream>>>(x, Wk, bk, Kw, DMODEL, DMODEL, 1);
  k_gemm_wmma<<<gridG, 256, 0, stream>>>(x, Wv, bv, Vw, DMODEL, DMODEL, 1);
  k_sample<<<(B_SZ * H_HEADS * L_SEQ) / 256, 256, 0, stream>>>(Qw, Kw, sidx, Mstat);
  k_topk<<<B_SZ * H_HEADS, 256, 0, stream>>>(Mstat, Mtop);
  k_attn<<<B_SZ * H_HEADS * U_TOP, 256, 0, stream>>>(Qw, Kw, Vw, Mtop, updw);
  k_cumsum<<<(B_SZ * DMODEL) / 256, 256, 0, stream>>>(Vw, ctx);
  k_scatter<<<(B_SZ * H_HEADS * U_TOP * D_HEAD + 255) / 256, 256, 0, stream>>>(Mtop, updw, ctx);
  k_gemm_wmma<<<gridG, 256, 0, stream>>>(ctx, Wo, bo, out, DMODEL, DMODEL, 0);
}